// RCCNLoss_81441169867202
// MI455X (gfx1250) — compile-verified
//
#include <hip/hip_runtime.h>
#include <hip/hip_bf16.h>

typedef float v2f __attribute__((ext_vector_type(2)));
typedef float v8f __attribute__((ext_vector_type(8)));
typedef unsigned int v4u __attribute__((ext_vector_type(4)));
typedef int v4i __attribute__((ext_vector_type(4)));
typedef int v8i __attribute__((ext_vector_type(8)));

#define NUM_CLS_P1    81                  // 80 fg classes + background
#define TILE_ELEMS    (16 * NUM_CLS_P1)   // 1296 floats, contiguous slab per tile
#define CHUNKS        20                  // WMMA covers cols 0..79; col 80 added scalar
#define ROWS_PER_TILE 16
#define WAVES_PER_BLOCK 8
#define BLOCK_THREADS 256
#define MAIN_BLOCKS   1024

#if __has_builtin(__builtin_amdgcn_tensor_load_to_lds)
#define USE_TDM 1
#else
#define USE_TDM 0
#endif

__global__ __launch_bounds__(BLOCK_THREADS)
void rcnn_loss_main(const float* __restrict__ cls_pred,
                    const float* __restrict__ bbox_pred,
                    const long long* __restrict__ cls_targets,
                    const float* __restrict__ bbox_targets,
                    float* __restrict__ blk_cls,
                    float* __restrict__ blk_reg,
                    unsigned int* __restrict__ blk_fg,
                    int N)
{
    __shared__ float lds[WAVES_PER_BLOCK][TILE_ELEMS];   // flat 16x81 tile per wave
    __shared__ float rowsum[WAVES_PER_BLOCK][ROWS_PER_TILE];
    __shared__ float wave_cls[WAVES_PER_BLOCK];
    __shared__ float wave_reg[WAVES_PER_BLOCK];
    __shared__ unsigned int wave_fg[WAVES_PER_BLOCK];

    const int lane  = threadIdx.x & 31;
    const int wave  = threadIdx.x >> 5;
    const int row16 = lane & 15;     // matrix row M this lane services
    const int half  = lane >> 4;     // 0: lanes 0-15, 1: lanes 16-31

    const int numTiles   = (N + ROWS_PER_TILE - 1) / ROWS_PER_TILE;
    const int totalWaves = gridDim.x * WAVES_PER_BLOCK;
    const int waveGlobal = blockIdx.x * WAVES_PER_BLOCK + wave;
    const int iters      = (numTiles + totalWaves - 1) / totalWaves;

    float clsAcc = 0.0f;
    float regAcc = 0.0f;
    unsigned int fgAcc = 0u;

    float* tile = lds[wave];

    for (int it = 0; it < iters; ++it) {
        const int  tileIdx  = waveGlobal + it * totalWaves;   // wave-uniform
        const int  tileCl   = (tileIdx < numTiles) ? tileIdx : (numTiles - 1); // clamp: always valid memory
        const int  row      = tileIdx * ROWS_PER_TILE + row16;
        const bool rowValid = (tileIdx < numTiles) && (row < N);

        // ---- Phase 0: bring the contiguous 5184B tile slab into LDS.
        // Previous iteration's DS reads must drain before TDM rewrites the buffer
        // (tensor ops are unordered vs DS ops per ISA).
        __syncthreads();
        asm volatile("s_wait_dscnt 0x0" ::: "memory");

#if USE_TDM
        {
            const unsigned long long ga =
                (unsigned long long)(uintptr_t)(cls_pred + (long long)tileCl * TILE_ELEMS);
            const unsigned ldsAddr = (unsigned)(uintptr_t)&tile[0];  // low 32 bits = LDS offset

            v4u g0;
            g0.x = 1u;                                   // count=1, is_restore=0, gather off
            g0.y = ldsAddr;                              // lds_addr[31:0]
            g0.z = (unsigned)(ga & 0xFFFFFFFFull);       // global_addr[31:0]
            g0.w = (unsigned)((ga >> 32) & 0x01FFFFFFull) | 0x80000000u; // addr[56:32] | type=2

            v8i g1;
            g1[0] = (int)(2u << 16);                     // wg_mask=0, data_size=2 (4B), no opts
            g1[1] = (int)((TILE_ELEMS & 0xFFFF) << 16);  // tensor_dim0[15:0] (low half @ bits 79:48)
            g1[2] = (int)((TILE_ELEMS >> 16) | (1u << 16));      // tensor_dim0[31:16] | tensor_dim1[15:0]=1
            g1[3] = (int)((0u) | ((unsigned)TILE_ELEMS << 16));  // tensor_dim1[31:16]=0 | tile_dim0=1296
            g1[4] = 1;                                   // tile_dim1=1, tile_dim2=0
            g1[5] = TILE_ELEMS;                          // tensor_dim0_stride[31:0]
            g1[6] = 0;                                   // stride0[47:32]=0, stride1[15:0]=0
            g1[7] = 0;                                   // stride1[47:16]=0

            v4i gz  = {0, 0, 0, 0};
            v8i gz8 = {0, 0, 0, 0, 0, 0, 0, 0};
            // clang-23 / therock-10.0 lane: 6-arg form (extra int32x8 before cpol)
            __builtin_amdgcn_tensor_load_to_lds(g0, g1, gz, gz, gz8, 0);
        }
#else
        {
            // Fallback: fully-coalesced float4 slab copy (tile base is 16B-aligned).
            const float4* __restrict__ src =
                reinterpret_cast<const float4*>(cls_pred + (long long)tileCl * TILE_ELEMS);
            float4* __restrict__ dst = reinterpret_cast<float4*>(tile);
            #pragma unroll
            for (int j = 0; j < TILE_ELEMS / 4 / 32; ++j)     // 10 rounds of b128
                dst[j * 32 + lane] = src[j * 32 + lane];
            if (lane < (TILE_ELEMS / 4) % 32)                  // tail chunks 320..323
                dst[(TILE_ELEMS / 4 / 32) * 32 + lane] = src[(TILE_ELEMS / 4 / 32) * 32 + lane];
        }
#endif

        // ---- Overlap with the DMA: fetch target class, prefetch the 16B bbox gather
        int tgt = 0;
        if (rowValid && half == 0) {
            tgt = (int)cls_targets[row];
            if (tgt > 0) {
                __builtin_prefetch(bbox_pred + (long long)row * (4 * (NUM_CLS_P1 - 1))
                                             + (long long)(tgt - 1) * 4, 0, 1);
            }
        }

#if USE_TDM
        asm volatile("s_wait_tensorcnt 0x0" ::: "memory");
#endif
        __syncthreads();

        // ---- Phase A: per-row max (two half-lanes per row) + target logit
        float m, x_t = 0.0f;
        {
            float lmax = -3.4e38f;
            const int base = row16 * NUM_CLS_P1;
            for (int c = half; c < NUM_CLS_P1; c += 2)
                lmax = fmaxf(lmax, tile[base + c]);
            m = fmaxf(lmax, __shfl_xor(lmax, 16, 32));
            if (half == 0)
                x_t = tile[base + tgt];
        }
        __syncthreads();   // all raw reads (incl. x_t) complete before exp overwrite

        // ---- Phase B: exp(x - rowmax) in place
        {
            const int base = row16 * NUM_CLS_P1;
            for (int c = half; c < NUM_CLS_P1; c += 2) {
                const int i = base + c;
                tile[i] = __expf(tile[i] - m);
            }
        }
        __syncthreads();

        // ---- Phase C: row sums of cols 0..79 via V_WMMA_F32_16X16X4_F32, B = ones.
        // A layout (16x4 f32): lanes 0-15 hold K=0,K=1; lanes 16-31 hold K=2,K=3.
        // Unconditional => EXEC all-1s around every WMMA.
        {
            v8f acc = {0.f, 0.f, 0.f, 0.f, 0.f, 0.f, 0.f, 0.f};
            v2f bone; bone.x = 1.0f; bone.y = 1.0f;
            const int abase = row16 * NUM_CLS_P1 + 2 * half;
            #pragma unroll
            for (int k = 0; k < CHUNKS; ++k) {
                v2f a;
                a.x = tile[abase + 4 * k];
                a.y = tile[abase + 4 * k + 1];
                acc = __builtin_amdgcn_wmma_f32_16x16x4_f32(
                        false, a, false, bone, (short)0, acc, false, false);
            }
            // D layout: VGPR r = row r (lanes 0-15) / row r+8 (lanes 16-31); all cols equal
            if (lane == 0) {
                #pragma unroll
                for (int r = 0; r < 8; ++r) rowsum[wave][r] = acc[r];
            }
            if (lane == 16) {
                #pragma unroll
                for (int r = 0; r < 8; ++r) rowsum[wave][8 + r] = acc[r];
            }
        }
        __syncthreads();

        // ---- Phase D: per-row finalize (one lane per row)
        if (rowValid && half == 0) {
            const float S = rowsum[wave][row16] + tile[row16 * NUM_CLS_P1 + 80]; // + col 80
            clsAcc += m + __logf(S) - x_t;            // -log_softmax at target
            if (tgt > 0) {
                const float4 p  = *reinterpret_cast<const float4*>(
                    bbox_pred + (long long)row * (4 * (NUM_CLS_P1 - 1))
                              + (long long)(tgt - 1) * 4);
                const float4 t4 = *reinterpret_cast<const float4*>(
                    bbox_targets + (long long)row * 4);
                float s = 0.0f, d, ad;
                d = p.x - t4.x; ad = fabsf(d); s += (ad < 1.0f) ? 0.5f * d * d : ad - 0.5f;
                d = p.y - t4.y; ad = fabsf(d); s += (ad < 1.0f) ? 0.5f * d * d : ad - 0.5f;
                d = p.z - t4.z; ad = fabsf(d); s += (ad < 1.0f) ? 0.5f * d * d : ad - 0.5f;
                d = p.w - t4.w; ad = fabsf(d); s += (ad < 1.0f) ? 0.5f * d * d : ad - 0.5f;
                regAcc += s;
                fgAcc  += 1u;
            }
        }
    }
    __syncthreads();

    // ---- Wave reduction (lanes 16-31 hold zeros, harmless)
    for (int off = 16; off > 0; off >>= 1) {
        clsAcc += __shfl_down(clsAcc, off, 32);
        regAcc += __shfl_down(regAcc, off, 32);
        fgAcc  += __shfl_down(fgAcc,  off, 32);
    }
    if (lane == 0) {
        wave_cls[wave] = clsAcc;
        wave_reg[wave] = regAcc;
        wave_fg[wave]  = fgAcc;
    }
    __syncthreads();

    // ---- Block reduction in a fixed order (deterministic), one slot per block
    if (threadIdx.x == 0) {
        float c = 0.0f, r = 0.0f;
        unsigned int f = 0u;
        for (int w = 0; w < WAVES_PER_BLOCK; ++w) {
            c += wave_cls[w];
            r += wave_reg[w];
            f += wave_fg[w];
        }
        blk_cls[blockIdx.x] = c;
        blk_reg[blockIdx.x] = r;
        blk_fg[blockIdx.x]  = f;
    }
}

__global__ __launch_bounds__(256)
void rcnn_loss_finalize(const float* __restrict__ blk_cls,
                        const float* __restrict__ blk_reg,
                        const unsigned int* __restrict__ blk_fg,
                        int numBlocks, int N, float* __restrict__ out)
{
    __shared__ double sc[256];
    __shared__ double sr[256];
    __shared__ unsigned long long sf[256];

    const int t = threadIdx.x;
    double c = 0.0, r = 0.0;
    unsigned long long f = 0ull;
    for (int i = t; i < numBlocks; i += 256) {   // fixed traversal order
        c += (double)blk_cls[i];
        r += (double)blk_reg[i];
        f += (unsigned long long)blk_fg[i];
    }
    sc[t] = c; sr[t] = r; sf[t] = f;
    __syncthreads();
    for (int off = 128; off > 0; off >>= 1) {    // fixed tree -> deterministic
        if (t < off) {
            sc[t] += sc[t + off];
            sr[t] += sr[t + off];
            sf[t] += sf[t + off];
        }
        __syncthreads();
    }
    if (t == 0) {
        const double cls = sc[0] / (double)N;
        const unsigned long long nfg = sf[0];
        const double reg = (nfg > 0ull) ? (sr[0] / (double)nfg) : 0.0;
        out[0] = (float)(cls + reg);   // total_loss (LAMBDA_REG = 1)
        out[1] = (float)cls;           // cls_loss
        out[2] = (float)reg;           // reg_loss
    }
}

extern "C" void kernel_launch(void* const* d_in, const int* in_sizes, int n_in,
                              void* d_out, int out_size, void* d_ws, size_t ws_size,
                              hipStream_t stream)
{
    const float*     cls_pred     = (const float*)d_in[0];
    const float*     bbox_pred    = (const float*)d_in[1];
    const long long* cls_targets  = (const long long*)d_in[2];   // int64 targets
    const float*     bbox_targets = (const float*)d_in[3];
    float*           out          = (float*)d_out;

    const int N = in_sizes[2];   // cls_targets element count == N

    float*        blk_cls = (float*)d_ws;
    float*        blk_reg = blk_cls + MAIN_BLOCKS;
    unsigned int* blk_fg  = (unsigned int*)(blk_reg + MAIN_BLOCKS);

    rcnn_loss_main<<<MAIN_BLOCKS, BLOCK_THREADS, 0, stream>>>(
        cls_pred, bbox_pred, cls_targets, bbox_targets,
        blk_cls, blk_reg, blk_fg, N);

    rcnn_loss_finalize<<<1, 256, 0, stream>>>(
        blk_cls, blk_reg, blk_fg, MAIN_BLOCKS, N, out);
}